// GINConv2d_73169062855214
// MI455X (gfx1250) — compile-verified
//
#include <hip/hip_runtime.h>

typedef float v2f __attribute__((ext_vector_type(2)));
typedef float v4f __attribute__((ext_vector_type(4)));
typedef float v8f __attribute__((ext_vector_type(8)));

#define B_      4
#define C_IN_   192
#define C_OUT_  192
#define N_      16384
#define K_      16
#define G_      4
#define GIN_    48
#define GOUT_   48

// ---------------- Pass 1: gather + (1+eps)*x + sum_k x[idx] ----------------
// Block: (channel-block of 4, batch, N/2 node range). 512 threads.
// The 4 channel rows are CHANNEL-INTERLEAVED in LDS (float4 per node) so each
// random neighbor gather is a single ds_load_b128 serving all 4 channels:
//   per node: 17 ds_load_b128 + 16 shifts + 32 v_pk_add_f32 (vs 64 ds_b32).
#define CBLK   4
#define T1     512
#define NSPLIT 2

__global__ __launch_bounds__(T1)
void gin_gather(const float* __restrict__ x,
                const int*   __restrict__ edge,
                const float* __restrict__ eps,
                float*       __restrict__ h) {
  __shared__ v4f xint[N_];            // 262144 bytes: xint[n] = {c0,c1,c2,c3}
  const int c0     = blockIdx.x * CBLK;
  const int b      = blockIdx.y;
  const int nstart = blockIdx.z * (N_ / NSPLIT);
  const int tid    = threadIdx.x;

  // Fill: per channel, coalesced b128 global reads, component-scatter into
  // LDS (16-dword-stride bank conflicts on the b32 stores; fill is ~5% of the kernel).
  float* xf = (float*)xint;
  #pragma unroll
  for (int c = 0; c < CBLK; ++c) {
    const float4* src = (const float4*)(x + (size_t)(b * C_IN_ + c0 + c) * N_);
    for (int i = tid; i < N_ / 4; i += T1) {
      float4 v = src[i];
      xf[(i * 4 + 0) * 4 + c] = v.x;
      xf[(i * 4 + 1) * 4 + c] = v.y;
      xf[(i * 4 + 2) * 4 + c] = v.z;
      xf[(i * 4 + 3) * 4 + c] = v.w;
    }
  }
  __syncthreads();

  const float e1 = 1.0f + eps[0];
  const int* eb = edge + (size_t)b * N_ * K_;     // edge_index[0] slice

  for (int n = nstart + tid; n < nstart + N_ / NSPLIT; n += T1) {
    // 16 neighbor indices: 64B contiguous per thread, coalesced across lanes.
    const int4* ep = (const int4*)(eb + (size_t)n * K_);
    int4 q0 = ep[0], q1 = ep[1], q2 = ep[2], q3 = ep[3];
    const int id[16] = {q0.x,q0.y,q0.z,q0.w, q1.x,q1.y,q1.z,q1.w,
                        q2.x,q2.y,q2.z,q2.w, q3.x,q3.y,q3.z,q3.w};
    v4f s0 = {0.f, 0.f, 0.f, 0.f};
    v4f s1 = {0.f, 0.f, 0.f, 0.f};
    #pragma unroll
    for (int k = 0; k < 16; k += 2) {   // two chains -> pk_add ILP
      s0 += xint[id[k]];
      s1 += xint[id[k + 1]];
    }
    const v4f hv = e1 * xint[n] + (s0 + s1);
    const size_t o = (size_t)(b * C_IN_ + c0) * N_ + n;
    h[o              ] = hv.x;
    h[o + (size_t)N_ ] = hv.y;
    h[o + 2ul * N_   ] = hv.z;
    h[o + 3ul * N_   ] = hv.w;
  }
}

// ---------------- Pass 2: grouped GEMM via V_WMMA_F32_16X16X4_F32 ----------
// Block: (256-node tile, group, batch). 256 threads = 8 waves.
// Each wave owns two adjacent 16-col tiles (nt = 2w, 2w+1) and loops mt=0..2:
// every A-fragment (weights) feeds TWO independent WMMA accumulator chains.
// Reads h from d_out, writes the result in place over the same rows (all
// global reads complete before __syncthreads; stores only after).
#define TN2     256
#define T2      256
#define HSTRIDE 264   // 256 + 8 pad: lanes 0-15 / 16-31 hit disjoint banks

__global__ __launch_bounds__(T2)
void gin_gemm(float* __restrict__ hout,
              const float* __restrict__ w,
              const float* __restrict__ bias) {
  __shared__ float hs[GIN_ * HSTRIDE];   // ~50KB
  const int n0  = blockIdx.x * TN2;
  const int g   = blockIdx.y;
  const int b   = blockIdx.z;
  const int tid = threadIdx.x;

  // Load 48 x 256 h slab for this group/tile into LDS (float4, coalesced).
  for (int t = tid; t < GIN_ * (TN2 / 4); t += T2) {
    const int i = t >> 6;                // row (input channel within group)
    const int v = t & 63;                // float4 column
    float4 val = *(const float4*)(hout + (size_t)(b * C_IN_ + g * GIN_ + i) * N_
                                       + n0 + v * 4);
    *(float4*)(&hs[i * HSTRIDE + v * 4]) = val;
  }
  __syncthreads();

  const int lane = tid & 31;
  const int wave = tid >> 5;
  const int ln16 = lane & 15;   // M for A-frag, N for B-frag / C/D
  const int kp   = lane >> 4;   // 0: holds K,K+1 ; 1: holds K+2,K+3

  const int ntA = 2 * wave;     // two 16-wide column tiles per wave
  #pragma unroll
  for (int mt = 0; mt < 3; ++mt) {
    v8f acc0 = {0.f,0.f,0.f,0.f,0.f,0.f,0.f,0.f};
    v8f acc1 = {0.f,0.f,0.f,0.f,0.f,0.f,0.f,0.f};

    // A (16x4 f32): lane m holds W[m][kbase+2*kp], W[m][kbase+2*kp+1]
    const float* wrow = w + (size_t)(g * GOUT_ + mt * 16 + ln16) * GIN_ + 2 * kp;
    // B (4x16 f32): lane n holds h[kbase+2*kp][n], h[kbase+2*kp+1][n]
    const float* hpA = &hs[(2 * kp) * HSTRIDE + ntA * 16 + ln16];
    const float* hpB = hpA + 16;

    #pragma unroll
    for (int kk = 0; kk < GIN_ / 4; ++kk) {        // 12 K-steps of 4
      v2f a;  a.x  = wrow[kk * 4];            a.y  = wrow[kk * 4 + 1];
      v2f b0; b0.x = hpA[(kk * 4) * HSTRIDE]; b0.y = hpA[(kk * 4 + 1) * HSTRIDE];
      v2f b1; b1.x = hpB[(kk * 4) * HSTRIDE]; b1.y = hpB[(kk * 4 + 1) * HSTRIDE];
      acc0 = __builtin_amdgcn_wmma_f32_16x16x4_f32(
                 false, a, false, b0, (short)0, acc0, false, false);
      acc1 = __builtin_amdgcn_wmma_f32_16x16x4_f32(
                 false, a, false, b1, (short)0, acc1, false, false);
    }

    // Epilogue: bias + ReLU, store in place. C/D layout: VGPR r ->
    // M = r (+8 for lanes 16-31), N = lane&15.
    #pragma unroll
    for (int r = 0; r < 8; ++r) {
      const int co = g * GOUT_ + mt * 16 + r + kp * 8;
      const float bv = bias[co];
      const size_t rowo = (size_t)(b * C_OUT_ + co) * N_;
      float v0 = acc0[r] + bv;  v0 = v0 > 0.f ? v0 : 0.f;
      float v1 = acc1[r] + bv;  v1 = v1 > 0.f ? v1 : 0.f;
      hout[rowo + n0 + (ntA    ) * 16 + ln16] = v0;
      hout[rowo + n0 + (ntA + 1) * 16 + ln16] = v1;
    }
  }
}

extern "C" void kernel_launch(void* const* d_in, const int* in_sizes, int n_in,
                              void* d_out, int out_size, void* d_ws, size_t ws_size,
                              hipStream_t stream) {
  (void)in_sizes; (void)n_in; (void)d_ws; (void)ws_size; (void)out_size;
  const float* x    = (const float*)d_in[0];
  const int*   edge = (const int*)  d_in[1];   // edge_index, use [0] half
  const float* w    = (const float*)d_in[2];
  const float* bias = (const float*)d_in[3];
  const float* eps  = (const float*)d_in[4];
  float* out = (float*)d_out;

  dim3 g1(C_IN_ / CBLK, B_, NSPLIT);            // 48 x 4 x 2 = 384 blocks
  gin_gather<<<g1, T1, 0, stream>>>(x, edge, eps, out);

  dim3 g2(N_ / TN2, G_, B_);                    // 64 x 4 x 4 = 1024 blocks
  gin_gemm<<<g2, T2, 0, stream>>>(out, w, bias);
}